// GCN_22625887715699
// MI455X (gfx1250) — compile-verified
//
#include <hip/hip_runtime.h>

typedef __attribute__((ext_vector_type(2))) float v2f;
typedef __attribute__((ext_vector_type(8))) float v8f;

#define NODES 50000

// ---------------------------------------------------------------------------
// GEMM: Y[M x N] = X[M x K] * W[K x N], fp32 via V_WMMA_F32_16X16X4_F32.
// One wave per 16-row stripe, covering all N/16 tiles (A fragment reused).
// A 16x4 f32 layout: lanes 0-15 -> {K0,K1}, lanes 16-31 -> {K2,K3} (per lane 2 VGPRs)
// B 4x16 f32 layout: VGPR0 = row K(2*half), VGPR1 = row K(2*half+1), N = lane&15
// C/D 16x16 f32: VGPR i -> M = i + 8*(lane>>4), N = lane&15
// ---------------------------------------------------------------------------
template <int K, int N>
__global__ void gcn_gemm_wmma_f32(const float* __restrict__ X,
                                  const float* __restrict__ W,
                                  float* __restrict__ Y, int M) {
    constexpr int NT = N / 16;
    const int wave = blockIdx.x * (blockDim.x >> 5) + (threadIdx.x >> 5);
    const int m0 = wave * 16;
    if (m0 >= M) return;                       // wave-uniform: EXEC stays all-1s
    const int lane = threadIdx.x & 31;
    const int half = lane >> 4;                // 0 or 1
    const int sub  = lane & 15;

    v8f acc[NT] = {};

    for (int k = 0; k < K; k += 4) {
        // A fragment: X[m0+sub][k + 2*half .. +1]  (8B-aligned contiguous pair)
        const v2f a = *(const v2f*)(X + (size_t)(m0 + sub) * K + k + 2 * half);
#pragma unroll
        for (int t = 0; t < NT; ++t) {
            v2f b;
            b[0] = W[(size_t)(k + 2 * half)     * N + t * 16 + sub];
            b[1] = W[(size_t)(k + 2 * half + 1) * N + t * 16 + sub];
            acc[t] = __builtin_amdgcn_wmma_f32_16x16x4_f32(
                false, a, false, b, (short)0, acc[t], false, false);
        }
    }

#pragma unroll
    for (int t = 0; t < NT; ++t) {
#pragma unroll
        for (int i = 0; i < 8; ++i) {
            Y[(size_t)(m0 + i + 8 * half) * N + t * 16 + sub] = acc[t][i];
        }
    }
}

// ---------------------------------------------------------------------------
// Weighted in-degree over real edges (self-loop handled analytically as +1).
// ---------------------------------------------------------------------------
__global__ void gcn_deg_kernel(const int* __restrict__ col,
                               const float* __restrict__ w,
                               float* __restrict__ deg, int E) {
    int i = blockIdx.x * blockDim.x + threadIdx.x;
    if (i < E) unsafeAtomicAdd(&deg[col[i]], w[i]);
}

__global__ void gcn_dinv_kernel(const float* __restrict__ deg,
                                float* __restrict__ dinv, int n) {
    int i = blockIdx.x * blockDim.x + threadIdx.x;
    if (i < n) dinv[i] = rsqrtf(deg[i] + 1.0f);   // self-loop weight 1 => deg > 0
}

// ---------------------------------------------------------------------------
// Edge aggregation: dst[col] += dinv[row]*w*dinv[col] * src[row].
// One wave per edge (incl. virtual self-loop edges e >= E), D/32 floats/lane,
// hardware global_atomic_add_f32 via unsafeAtomicAdd.
// ---------------------------------------------------------------------------
template <int D>
__global__ void gcn_scatter_agg(const int* __restrict__ row,
                                const int* __restrict__ col,
                                const float* __restrict__ w,
                                const float* __restrict__ dinv,
                                const float* __restrict__ src,
                                float* __restrict__ dst, int E, int n) {
    const int wave = blockIdx.x * (blockDim.x >> 5) + (threadIdx.x >> 5);
    const int lane = threadIdx.x & 31;
    if (wave >= E + n) return;
    int r, c; float wt;
    if (wave < E) { r = row[wave]; c = col[wave]; wt = w[wave]; }
    else          { r = c = wave - E; wt = 1.0f; }
    const float nrm = dinv[r] * wt * dinv[c];
    const float* s = src + (size_t)r * D;
    float*       d = dst + (size_t)c * D;
#pragma unroll
    for (int j = lane; j < D; j += 32)
        unsafeAtomicAdd(&d[j], nrm * s[j]);
}

__global__ void gcn_bias_relu(const float* __restrict__ a,
                              const float* __restrict__ b,
                              float* __restrict__ h, int total, int D) {
    int i = blockIdx.x * blockDim.x + threadIdx.x;
    if (i < total) h[i] = fmaxf(a[i] + b[i % D], 0.0f);
}

// ---------------------------------------------------------------------------
// out = log_softmax(agg + b) over 64 features; one wave per node, 2 vals/lane.
// ---------------------------------------------------------------------------
__global__ void gcn_logsoftmax64(const float* __restrict__ agg,
                                 const float* __restrict__ b,
                                 float* __restrict__ out, int n) {
    const int wave = blockIdx.x * (blockDim.x >> 5) + (threadIdx.x >> 5);
    const int lane = threadIdx.x & 31;
    if (wave >= n) return;
    const size_t base = (size_t)wave * 64;
    float v0 = agg[base + lane]      + b[lane];
    float v1 = agg[base + 32 + lane] + b[32 + lane];
    float m = fmaxf(v0, v1);
#pragma unroll
    for (int off = 16; off > 0; off >>= 1)
        m = fmaxf(m, __shfl_xor(m, off, 32));
    float s = __expf(v0 - m) + __expf(v1 - m);
#pragma unroll
    for (int off = 16; off > 0; off >>= 1)
        s += __shfl_xor(s, off, 32);
    const float ls = m + __logf(s);
    out[base + lane]      = v0 - ls;
    out[base + 32 + lane] = v1 - ls;
}

extern "C" void kernel_launch(void* const* d_in, const int* in_sizes, int n_in,
                              void* d_out, int out_size, void* d_ws, size_t ws_size,
                              hipStream_t stream) {
    const float* x  = (const float*)d_in[0];   // [N, 256]
    const int*   ei = (const int*)d_in[1];     // [2, E]
    const float* ew = (const float*)d_in[2];   // [E]
    const float* W1 = (const float*)d_in[3];   // [256, 128]
    const float* b1 = (const float*)d_in[4];   // [128]
    const float* W2 = (const float*)d_in[5];   // [128, 64]
    const float* b2 = (const float*)d_in[6];   // [64]
    float* out = (float*)d_out;                // [N, 64]

    const int n = NODES;
    const int E = in_sizes[2];
    const int* row = ei;
    const int* col = ei + E;

    // Workspace carve-up (floats)
    float* deg  = (float*)d_ws;                       // n
    float* dinv = deg  + n;                           // n
    float* xw1  = dinv + n;                           // n*128 (also reused as h)
    float* agg1 = xw1  + (size_t)n * 128;             // n*128
    float* xw2  = agg1 + (size_t)n * 128;             // n*64
    float* agg2 = xw2  + (size_t)n * 64;              // n*64

    hipMemsetAsync(deg,  0, (size_t)n * sizeof(float), stream);
    hipMemsetAsync(agg1, 0, (size_t)n * 128 * sizeof(float), stream);
    hipMemsetAsync(agg2, 0, (size_t)n * 64  * sizeof(float), stream);

    // Degree + normalization
    gcn_deg_kernel<<<(E + 255) / 256, 256, 0, stream>>>(col, ew, deg, E);
    gcn_dinv_kernel<<<(n + 255) / 256, 256, 0, stream>>>(deg, dinv, n);

    const int mWaves  = n / 16;                      // 3125 (exact)
    const int gemmBlk = (mWaves + 3) / 4;            // 4 waves / 128-thread block
    const int eWaves  = E + n;
    const int edgeBlk = (eWaves + 7) / 8;            // 8 waves / 256-thread block

    // Layer 1
    gcn_gemm_wmma_f32<256, 128><<<gemmBlk, 128, 0, stream>>>(x, W1, xw1, n);
    gcn_scatter_agg<128><<<edgeBlk, 256, 0, stream>>>(row, col, ew, dinv, xw1, agg1, E, n);
    gcn_bias_relu<<<((size_t)n * 128 + 255) / 256, 256, 0, stream>>>(agg1, b1, xw1, n * 128, 128);

    // Layer 2
    gcn_gemm_wmma_f32<128, 64><<<gemmBlk, 128, 0, stream>>>(xw1, W2, xw2, n);
    gcn_scatter_agg<64><<<edgeBlk, 256, 0, stream>>>(row, col, ew, dinv, xw2, agg2, E, n);

    // Output
    gcn_logsoftmax64<<<(n + 7) / 8, 256, 0, stream>>>(agg2, b2, out, n);
}